// CHILDREN_TENSOR_75737453297667
// MI455X (gfx1250) — compile-verified
//
#include <hip/hip_runtime.h>
#include <hip/hip_bf16.h>

// Pure gather: out[b,n,c,:] = (children[b,n,c]==0) ? 0 : nodes[b, children[b,n,c], :]
// B=16, N=2048, C=32, F=128  ->  rows = B*N*C = 1<<20, 512 B per row.
//
// Roofline: 512 MB NT stores + ~20 MB reads @ 23.3 TB/s => ~23 us floor.
//
// Layout: each wave32 owns 32 consecutive rows (16 KB of output).
//  - one coalesced b32 load grabs the wave's 32 indices (one per lane)
//  - v_readlane moves each index to an SGPR -> node-row address is scalar math,
//    gather uses SADDR-form global_load_b128 with loop-invariant lane*16 voffset
//  - stores are SADDR-form NT b128 with r*512 folded into the ioffset immediate
// Steady state ~8-10 issue slots per 512 B row, so instruction issue can never
// contend with the 23.3 TB/s store stream.

typedef float v4f __attribute__((ext_vector_type(4)));

__global__ __launch_bounds__(256)
void CHILDREN_TENSOR_gather_kernel(const float* __restrict__ nodes,
                                   const int* __restrict__ children,
                                   float* __restrict__ out)
{
    const unsigned wave    = (blockIdx.x * 256u + threadIdx.x) >> 5;
    const unsigned lane    = threadIdx.x & 31u;
    const unsigned rowBase = wave << 5;            // 32 rows per wave

    // All 32 rows of this wave share one batch: b = rowBase >> 16 (N*C = 65536,
    // and 32 | 65536). Per-batch nodes stride = N*F = 1<<18 floats.
    const unsigned b = rowBase >> 16;
    const float* __restrict__ nb = nodes + ((size_t)b << 18);   // batch base
    float*       __restrict__ ob = out   + ((size_t)rowBase << 7); // wave out base

    // One coalesced load: lane L holds the index for row rowBase+L.
    const int idxv = children[rowBase + lane];

    #pragma unroll 4
    for (int r = 0; r < 32; ++r) {
        const int idx = __builtin_amdgcn_readlane(idxv, r);  // wave-uniform SGPR

        // Row 0 is valid memory: load unconditionally, select zero after.
        const v4f* __restrict__ src = (const v4f*)(nb + ((unsigned)idx << 7));
        v4f v = src[lane];
        if (idx == 0) {
            v = (v4f)0.0f;
        }

        // Write-once 512 MB stream: non-temporal so it can't evict the 16 MB
        // nodes table from the 192 MB L2 (keeps the gather reads L2-resident).
        v4f* dst = (v4f*)(ob + ((unsigned)r << 7)) + lane;
        __builtin_nontemporal_store(v, dst);
    }
}

extern "C" void kernel_launch(void* const* d_in, const int* in_sizes, int n_in,
                              void* d_out, int out_size, void* d_ws, size_t ws_size,
                              hipStream_t stream) {
    (void)in_sizes; (void)n_in; (void)d_ws; (void)ws_size; (void)out_size;

    const float* nodes    = (const float*)d_in[0];  // [16, 2048, 128] f32
    const int*   children = (const int*)  d_in[1];  // [16, 2048, 32]  i32
    // d_in[2] is the feature_size scalar (==128); compile-time constant here.
    float* out = (float*)d_out;                     // [16, 2048, 32, 128] f32

    const unsigned total_rows     = 16u * 2048u * 32u; // 1<<20
    const unsigned rows_per_wave  = 32;
    const unsigned waves_per_blk  = 8;                 // 256 threads
    const unsigned rows_per_block = rows_per_wave * waves_per_blk; // 256
    dim3 grid(total_rows / rows_per_block);            // 4096 blocks
    dim3 block(256);

    CHILDREN_TENSOR_gather_kernel<<<grid, block, 0, stream>>>(nodes, children, out);
}